// PatchCCADense_12661563589194
// MI455X (gfx1250) — compile-verified
//
#include <hip/hip_runtime.h>
#include <hip/hip_bf16.h>
#include <math.h>

// ---------------------------------------------------------------------------
// PatchCCA on MI455X (gfx1250, wave32).
//
// Per (b,h,w,scale) site: Z = [x_patch | y_patch]  (n x 16, n = k*k).
// G = Z^T Z (16x16) via chained V_WMMA_F32_16X16X4_F32 (exact f32),
// colsums via a second WMMA with A = ones. Standardized covariances derived
// from G + colsums, then 8x8 Cholesky / tri-inverse / diag contraction.
//
// v2: block-level LDS tile (zero-padded, plus one extra all-zero patch-row)
// makes every fragment gather a branchless ds_load_b32; all NCH fragments are
// preloaded so the WMMA chain runs back-to-back after one LDS wait.
//
// Fragment layout (CDNA5 ISA 7.12.2, 32-bit A 16x4 / B 4x16, wave32):
//   lane half hi = lane>>4, col = lane&15, VGPR v in {0,1}:
//     A[m=col][k = v + 2*hi]  and  B[k = v + 2*hi][n=col]
//   With A = Z^T and B = Z both fragments equal Z[k0 + v + 2*hi][col],
//   so a single v2f register pair feeds both WMMA operands.
// ---------------------------------------------------------------------------

typedef __attribute__((ext_vector_type(2))) float v2f;
typedef __attribute__((ext_vector_type(8))) float v8f;

#define CCA_EPS 1e-6f

static constexpr int Bc = 4, Cc = 8, Hc = 256, Wc = 256, Lc = Hc * Wc;
static constexpr int WAVES_PER_BLOCK = 8;
static constexpr int SITES_PER_BLOCK = 64;   // one 64-wide row segment
static constexpr int SITES_PER_WAVE  = SITES_PER_BLOCK / WAVES_PER_BLOCK; // 8
static constexpr int NBLOCKS = (Bc * Lc) / SITES_PER_BLOCK;               // 4096

// Per-wave tail scratch layout (floats)
static constexpr int WS_G   = 0;     // 16x16 Gram            (256)
static constexpr int WS_CS  = 256;   // colsums               (16)
static constexpr int WS_MU  = 272;   // means                 (16)
static constexpr int WS_SD  = 288;   // sample stddevs        (16)
static constexpr int WS_AXX = 304;   // cov_xx 8x8            (64)
static constexpr int WS_AYY = 368;   // cov_yy 8x8            (64)
static constexpr int WS_CXY = 432;   // cov_xy 8x8            (64)
static constexpr int WS_MX  = 496;   // Lx^-1 packed lower    (36)
static constexpr int WS_MY  = 532;   // Ly^-1 packed lower    (36)
static constexpr int WS_STRIDE = 576;

template <int K, int PAD>
__global__ __launch_bounds__(256) void patchcca_kernel(
    const float* __restrict__ x, const float* __restrict__ y,
    float* __restrict__ out, int scaleIdx)
{
    constexpr int N    = K * K;                 // patch rows (9 / 25)
    constexpr int NCH  = (N + 3) / 4;           // WMMA K=4 chunks (3 / 7)
    constexpr int TW   = SITES_PER_BLOCK + 2 * PAD;   // tile width
    constexpr int IDIM = K + 1;                 // patch rows + 1 zero row
    constexpr int CHS  = IDIM * TW + 1;         // channel stride (odd -> no
                                                //  LDS bank clash across lanes)
    constexpr int TILE = 16 * CHS;              // tile floats

    __shared__ float smem[TILE + WAVES_PER_BLOCK * WS_STRIDE];
    float* tile = smem;

    const int tid  = threadIdx.x;
    const int lane = tid & 31;
    const int wib  = tid >> 5;
    const int hi   = lane >> 4;   // lane half
    const int col  = lane & 15;   // Z column (channel) owned by this lane

    // Block -> (b, h, w0): 4 segments of 64 sites per image row.
    const int row = blockIdx.x >> 2;       // 0..1023
    const int seg = blockIdx.x & 3;
    const int b   = row >> 8;
    const int h   = row & 255;
    const int w0  = seg << 6;

    // ---- cooperative tile fill: tile[c][i][tc], zero-padded everywhere ----
    // tc maps to global column (w0 - PAD + tc); row i maps to (h + i - PAD);
    // row i == K (and the 1-float channel pad) stay zero.
    for (int e = tid; e < TILE; e += 256) {
        const int c   = e / CHS;
        const int rem = e - c * CHS;
        const int i   = rem / TW;
        const int tc  = rem - i * TW;
        float val = 0.0f;
        if (i < K) {
            const int hh = h + i - PAD;
            const int gw = w0 - PAD + tc;
            if (hh >= 0 && hh < Hc && gw >= 0 && gw < Wc) {
                const float* src = (c < 8) ? x : y;
                val = src[(size_t)((b * Cc + (c & 7)) * Hc + hh) * Wc + gw];
            }
        }
        tile[e] = val;
    }
    __syncthreads();

    // ---- per-wave tail scratch ----
    float* ws  = smem + TILE + wib * WS_STRIDE;
    float* G   = ws + WS_G;
    float* CS  = ws + WS_CS;
    float* MU  = ws + WS_MU;
    float* SD  = ws + WS_SD;
    float* AXX = ws + WS_AXX;
    float* AYY = ws + WS_AYY;
    float* CXY = ws + WS_CXY;
    float* MX  = ws + WS_MX;
    float* MY  = ws + WS_MY;

    const float* myCh = tile + col * CHS;   // this lane's channel plane

    for (int t = 0; t < SITES_PER_WAVE; ++t) {
        const int wl = wib * SITES_PER_WAVE + t;   // site within segment
        const int w  = w0 + wl;
        const size_t p = (size_t)(b * Lc + h * Wc + w);

        // ---- branchless fragment gathers (all chunks up front) ----
        v2f frag[NCH];
        #pragma unroll
        for (int q = 0; q < NCH; ++q) {
            #pragma unroll
            for (int v = 0; v < 2; ++v) {
                const int r = q * 4 + v + 2 * hi;   // Z row; r >= N hits zero row
                const int i = r / K;
                const int j = r - i * K;
                frag[q][v] = myCh[i * TW + wl + j];
            }
        }

        // ---- WMMA chain: G += Zc^T Zc ; colsums += ones^T Zc ----
        v8f acc  = {0.f, 0.f, 0.f, 0.f, 0.f, 0.f, 0.f, 0.f};
        v8f accS = {0.f, 0.f, 0.f, 0.f, 0.f, 0.f, 0.f, 0.f};
        v2f ones; ones.x = 1.0f; ones.y = 1.0f;
        #pragma unroll
        for (int q = 0; q < NCH; ++q) {
            acc  = __builtin_amdgcn_wmma_f32_16x16x4_f32(
                       false, frag[q], false, frag[q], (short)0, acc, false, false);
            accS = __builtin_amdgcn_wmma_f32_16x16x4_f32(
                       false, ones, false, frag[q], (short)0, accS, false, false);
        }

        // ---- dump G (C/D layout: VGPR r -> row r + 8*hi, lanes -> col) ----
        #pragma unroll
        for (int vg = 0; vg < 8; ++vg)
            G[(vg + 8 * hi) * 16 + col] = acc[vg];
        if (hi == 0) CS[col] = accS[0];   // row 0 of ones^T Z = colsum[col]
        __syncthreads();

        // ---- stage 1: per-column mean / sample stddev ----
        if (hi == 0) {
            const float mu  = CS[col] * (1.0f / (float)N);
            const float var = (G[col * 16 + col] - (float)N * mu * mu)
                              * (1.0f / (float)(N - 1));
            MU[col] = mu;
            SD[col] = sqrtf(var);
        }
        __syncthreads();

        // ---- stage 2: standardized covariances (192 entries / 32 lanes) ----
        #pragma unroll
        for (int e0 = 0; e0 < 6; ++e0) {
            const int e  = lane + e0 * 32;       // 0..191
            const int m  = e >> 6;               // 0=cxx 1=cyy 2=cxy
            const int r8 = (e >> 3) & 7;
            const int c8 = e & 7;
            const int a  = r8 + ((m == 1) ? 8 : 0);
            const int bb = c8 + ((m >= 1) ? 8 : 0);
            float cov = (G[a * 16 + bb] - (float)N * MU[a] * MU[bb])
                        / ((float)(N - 1) * SD[a] * SD[bb]);
            if (m < 2 && r8 == c8) cov += CCA_EPS;
            float* dst = (m == 0) ? AXX : (m == 1) ? AYY : CXY;
            dst[r8 * 8 + c8] = cov;
        }
        __syncthreads();

        // ---- stage 3: Cholesky + lower-tri inverse (lane0: xx, lane1: yy) ----
        if (lane < 2) {
            const float* A = (lane == 0) ? AXX : AYY;
            float*       M = (lane == 0) ? MX  : MY;
            float Llo[36];   // packed lower: (i,j) -> i*(i+1)/2 + j
            #pragma unroll
            for (int j = 0; j < 8; ++j) {
                float d = A[j * 8 + j];
                #pragma unroll
                for (int t2 = 0; t2 < j; ++t2) {
                    const float v = Llo[j * (j + 1) / 2 + t2];
                    d -= v * v;
                }
                const float lj  = sqrtf(d);
                const float inv = 1.0f / lj;
                Llo[j * (j + 1) / 2 + j] = lj;
                #pragma unroll
                for (int i = 0; i < 8; ++i) {
                    if (i > j) {
                        float s = A[i * 8 + j];
                        #pragma unroll
                        for (int t2 = 0; t2 < j; ++t2)
                            s -= Llo[i * (i + 1) / 2 + t2] * Llo[j * (j + 1) / 2 + t2];
                        Llo[i * (i + 1) / 2 + j] = s * inv;
                    }
                }
            }
            float Mi[36];
            #pragma unroll
            for (int i = 0; i < 8; ++i) {
                const float dinv = 1.0f / Llo[i * (i + 1) / 2 + i];
                Mi[i * (i + 1) / 2 + i] = dinv;
                #pragma unroll
                for (int j = 0; j < 8; ++j) {
                    if (j < i) {
                        float s = 0.0f;
                        #pragma unroll
                        for (int t2 = 0; t2 < 8; ++t2)
                            if (t2 >= j && t2 < i)
                                s += Llo[i * (i + 1) / 2 + t2] * Mi[t2 * (t2 + 1) / 2 + j];
                        Mi[i * (i + 1) / 2 + j] = -dinv * s;
                    }
                }
            }
            #pragma unroll
            for (int q2 = 0; q2 < 36; ++q2) M[q2] = Mi[q2];
        }
        __syncthreads();

        // ---- stage 4: sum |diag(Lx^-1 Cxy Ly^-1)| / 8 ----
        if (lane == 0) {
            float accd = 0.0f;
            #pragma unroll
            for (int i = 0; i < 8; ++i) {
                float d = 0.0f;
                #pragma unroll
                for (int k2 = 0; k2 < 8; ++k2) {
                    if (k2 >= i) {
                        float tt = 0.0f;
                        #pragma unroll
                        for (int j = 0; j < 8; ++j)
                            if (j <= i)
                                tt += MX[i * (i + 1) / 2 + j] * CXY[j * 8 + k2];
                        d += tt * MY[k2 * (k2 + 1) / 2 + i];
                    }
                }
                accd += fabsf(d);
            }
            out[p * 2 + scaleIdx] = accd * 0.125f;
        }
        __syncthreads();   // protect per-wave scratch before next site
    }
}

extern "C" void kernel_launch(void* const* d_in, const int* in_sizes, int n_in,
                              void* d_out, int out_size, void* d_ws, size_t ws_size,
                              hipStream_t stream) {
    const float* x = (const float*)d_in[0];
    const float* y = (const float*)d_in[1];
    float* out = (float*)d_out;

    patchcca_kernel<3, 1><<<NBLOCKS, 256, 0, stream>>>(x, y, out, 0);
    patchcca_kernel<5, 2><<<NBLOCKS, 256, 0, stream>>>(x, y, out, 1);
}